// StableFactor_13280038880162
// MI455X (gfx1250) — compile-verified
//
#include <hip/hip_runtime.h>

// out[i,j] = alpha[i] * alpha[j] * A[i,j], N = 10000, fp32.
// HBM-bound: 800 MB/call -> ~34 us floor @ 23.3 TB/s. WMMA is intentionally
// unused (elementwise op, 0.25 FLOP/byte; matrix pipes can't move the floor).
// Path the chip rewards: b128 non-temporal loads/stores for the streamed-once
// 400 MB A/out arrays, cached (RT) loads for the 40 KB alpha vector, one
// gfx1250 global_prefetch per thread, straight-line fully-unrolled body.

typedef __attribute__((ext_vector_type(4))) float v4f;

#define NN    10000
#define VPR   (NN / 4)        // 2500 float4 per row (40000 B row stride, 16B aligned)
#define TPB   256             // 8 wave32 per block
#define FULL  (VPR / TPB)     // 9 guaranteed iterations per thread
#define TAIL  (VPR - FULL*TPB)// 196 tail lanes

__global__ __launch_bounds__(TPB) void diag_scale_kernel(
    const float* __restrict__ A,
    const float* __restrict__ alpha,
    float* __restrict__ out)
{
    const int row = blockIdx.x;
    const int tid = threadIdx.x;
    const float ai = alpha[row];                        // uniform -> s_load

    const v4f* __restrict__ arow = (const v4f*)(A   + (size_t)row * NN);
    v4f*       __restrict__ orow = (v4f*)      (out + (size_t)row * NN);
    const v4f* __restrict__ al4  = (const v4f*)alpha;   // RT cached path

    // gfx1250 prefetch (global_prefetch_b8) for the tail region; guarded so
    // the address stays inside A (worst case: start of the next row).
    if (row + 1 < NN)
        __builtin_prefetch((const void*)(arow + FULL * TPB + tid), 0, 0);

    // 9 guaranteed iterations: fully unrolled straight-line stream.
    // Independent NT b128 loads issue back-to-back -> deep MLP per wave.
    #pragma unroll
    for (int k = 0; k < FULL; ++k) {
        const int jv = k * TPB + tid;
        v4f a  = __builtin_nontemporal_load(arow + jv); // th:NT (streamed once)
        v4f aj = al4[jv];                               // tiny, keep cached
        v4f r  = (ai * aj) * a;
        __builtin_nontemporal_store(r, orow + jv);      // th:NT
    }

    // Tail: lanes 0..TAIL-1 handle the last 196 float4 of the row.
    if (tid < TAIL) {
        const int jv = FULL * TPB + tid;
        v4f a  = __builtin_nontemporal_load(arow + jv);
        v4f aj = al4[jv];
        v4f r  = (ai * aj) * a;
        __builtin_nontemporal_store(r, orow + jv);
    }
}

extern "C" void kernel_launch(void* const* d_in, const int* in_sizes, int n_in,
                              void* d_out, int out_size, void* d_ws, size_t ws_size,
                              hipStream_t stream)
{
    const float* A     = (const float*)d_in[0];   // (N, N) fp32
    const float* alpha = (const float*)d_in[1];   // (N,)  fp32
    float* out         = (float*)d_out;           // (N, N) fp32

    (void)in_sizes; (void)n_in; (void)out_size; (void)d_ws; (void)ws_size;

    diag_scale_kernel<<<dim3(NN), dim3(TPB), 0, stream>>>(A, alpha, out);
}